// HarmonicOscillator_88759794139745
// MI455X (gfx1250) — compile-verified
//
#include <hip/hip_runtime.h>
#include <math.h>
#include <stdint.h>

#define NH   32
#define SEG  960
#define LF   200
#define LW   (LF * SEG)          // 192000
#define SRD  48000.0

// Reciprocals as compile-time constants: avoids v_div_scale_f64/v_div_fmas_f64
// chains (the compiler cannot strength-reduce f64 division by 46080000).
#define INV_SR        (1.0 / 48000.0)
#define INV_SEG_SR    (1.0 / 46080000.0)     // 1/(960*48000)

// ---------------------------------------------------------------------------
// Closed-form coefficients of the piecewise-linear interpolated f0 within one
// 960-sample segment k.  Samples j=0..479 (piece 1) interpolate F[k-1]..F[k],
// samples j=480..959 (piece 2) interpolate F[k]..F[k+1].  Edge clipping at
// k==0 / k==LF-1 falls out naturally because the clamped neighbor makes the
// slope delta zero (matches jnp.clip(pos, 0, Lf-1)).
//   g1(j)  = a1 + j*b1   (f0_w / SR, piece 1)
//   g2(j') = a2 + j'*b2  (piece 2, j' = j-480)
// ---------------------------------------------------------------------------
__device__ __forceinline__ void seg_coeffs(const float* __restrict__ F, int k,
                                           double& a1, double& b1,
                                           double& a2, double& b2) {
    double Fm = (double)F[k > 0 ? k - 1 : 0];
    double Fc = (double)F[k];
    double Fp = (double)F[k < LF - 1 ? k + 1 : LF - 1];
    double d1 = Fc - Fm;
    double d2 = Fp - Fc;
    a1 = (Fm + (0.5 + 0.5 / 960.0) * d1) * INV_SR;
    b1 = d1 * INV_SEG_SR;
    a2 = (Fc + (0.5 / 960.0) * d2) * INV_SR;
    b2 = d2 * INV_SEG_SR;
}

// ---------------------------------------------------------------------------
// Kernel 1: per-batch exclusive prefix of per-segment phase integrals (f64).
// Segment total = 480*(a1+a2) + (479*480/2)*(b1+b2).  One block per batch,
// Hillis-Steele scan over 200 doubles in LDS.  Writes d_ws[n*LF + k].
// ---------------------------------------------------------------------------
__global__ void __launch_bounds__(256)
harm_prefix_kernel(const float* __restrict__ f0, double* __restrict__ pref) {
    const int n   = blockIdx.x;
    const int tid = threadIdx.x;
    const float* F = f0 + n * LF;

    __shared__ double sh[256];
    double T = 0.0;
    if (tid < LF) {
        double a1, b1, a2, b2;
        seg_coeffs(F, tid, a1, b1, a2, b2);
        T = 480.0 * (a1 + a2) + 114960.0 * (b1 + b2);   // 114960 = 479*480/2
    }
    sh[tid] = T;
    __syncthreads();
#pragma unroll
    for (int off = 1; off < 256; off <<= 1) {
        double v = (tid >= off) ? sh[tid - off] : 0.0;
        __syncthreads();
        sh[tid] += v;
        __syncthreads();
    }
    if (tid < LF) {
        double excl = (tid == 0) ? 0.0 : sh[tid - 1];
        pref[n * LF + tid] = excl;
    }
}

// ---------------------------------------------------------------------------
// Kernel 2: main synthesis.  256 threads = 8 waves, one sample per thread.
// Frame data for the block (<=3 frames x 32 harmonics) is staged to LDS with
// the gfx1250 async-to-LDS path (ASYNCcnt-tracked); phase is computed in f64
// closed form while the transfer is in flight, then reduced mod 1 (valid
// because the harmonic index h is an integer).
// ---------------------------------------------------------------------------
__global__ void __launch_bounds__(256)
harm_main_kernel(const float*  __restrict__ f0,
                 const float*  __restrict__ amps,
                 const double* __restrict__ pref,
                 float*        __restrict__ out) {
    const int n   = blockIdx.y;
    const int t0  = blockIdx.x * 256;
    const int tid = threadIdx.x;
    const int t   = t0 + tid;

    __shared__ float buf[NH * 3];   // [harmonic][frame - fi_lo]

    // Lowest frame index touched by this block (frame index is nondecreasing
    // in t and advances by <=1 over 256 samples -> frames fi_lo..fi_lo+2).
    const int k0 = t0 / SEG;
    const int s0 = t0 - k0 * SEG;
    const int fi_lo = (s0 >= 480) ? k0 : ((k0 > 0) ? k0 - 1 : 0);

    // --- async stage: 32 harmonics x 3 frames -> LDS (ASYNCcnt-tracked) ---
    if (tid < NH * 3) {
        const int h = tid / 3;
        const int f = tid - h * 3;
        int fr = fi_lo + f;
        if (fr > LF - 1) fr = LF - 1;
        unsigned goff   = (unsigned)((((n * NH + h) * LF) + fr) * 4);
        unsigned ldsoff = (unsigned)(uintptr_t)(&buf[h * 3 + f]);
        asm volatile("global_load_async_to_lds_b32 %0, %1, %2"
                     :: "v"(ldsoff), "v"(goff), "s"(amps) : "memory");
    }

    // --- closed-form phase integral while the async load is in flight ---
    const int k = t / SEG;
    const int s = t - k * SEG;
    const float* F = f0 + n * LF;
    double a1, b1, a2, b2;
    seg_coeffs(F, k, a1, b1, a2, b2);
    double P;
    if (s < 480) {
        double ds = (double)s;
        P = (ds + 1.0) * a1 + b1 * (ds * (ds + 1.0) * 0.5);
    } else {
        double T1 = 480.0 * a1 + 114960.0 * b1;
        double ds = (double)(s - 480);
        P = T1 + (ds + 1.0) * a2 + b2 * (ds * (ds + 1.0) * 0.5);
    }
    double Phi   = pref[n * LF + k] + P;         // inclusive cumsum of f0_w/SR
    double fracp = Phi - floor(Phi);             // sin(2*pi*h*Phi)=sin(2*pi*h*frac)
    float  theta = (float)(6.283185307179586 * fracp);

    // --- interpolation weights for amps (uniform per wave: 480 % 32 == 0) ---
    int fi0, fi1; float w;
    if (s >= 480) {
        if (k == LF - 1) { fi0 = LF - 1; fi1 = LF - 1; w = 0.0f; }
        else { fi0 = k; fi1 = k + 1; w = ((float)(s - 480) + 0.5f) * (1.0f / 960.0f); }
    } else {
        if (k == 0) { fi0 = 0; fi1 = 1; w = 0.0f; }
        else { fi0 = k - 1; fi1 = k; w = 0.5f + ((float)s + 0.5f) * (1.0f / 960.0f); }
    }
    const int ia = fi0 - fi_lo;
    const int ib = fi1 - fi_lo;

    float sth, cth;
    __sincosf(theta, &sth, &cth);
    const float c2x = 2.0f * cth;

    // wait for async LDS fill, then make it block-visible
    asm volatile("s_wait_asynccnt 0x0" ::: "memory");
    __syncthreads();

    // Chebyshev recurrence: sin((h+1)a) = 2cos(a)sin(ha) - sin((h-1)a).
    // 2 trans ops total instead of 32; LDS reads are wave-uniform broadcasts.
    float sprev = 0.0f, scur = sth, acc = 0.0f;
#pragma unroll
    for (int h = 0; h < NH; ++h) {
        float A   = buf[h * 3 + ia];
        float B   = buf[h * 3 + ib];
        float amp = fmaf(w, B - A, A);
        acc = fmaf(amp, scur, acc);
        float snext = fmaf(c2x, scur, -sprev);
        sprev = scur;
        scur  = snext;
    }

    out[n * LW + t] = acc * (1.0f / 32.0f);
}

// ---------------------------------------------------------------------------
extern "C" void kernel_launch(void* const* d_in, const int* in_sizes, int n_in,
                              void* d_out, int out_size, void* d_ws, size_t ws_size,
                              hipStream_t stream) {
    const float* f0   = (const float*)d_in[0];   // (N,1,200)
    const float* amps = (const float*)d_in[1];   // (N,32,200)
    float* out = (float*)d_out;                  // (N,1,192000)
    double* pref = (double*)d_ws;                // N*200 doubles of scratch

    const int N = in_sizes[0] / LF;              // 1600/200 = 8

    harm_prefix_kernel<<<dim3(N), dim3(256), 0, stream>>>(f0, pref);
    harm_main_kernel<<<dim3(LW / 256, N), dim3(256), 0, stream>>>(f0, amps, pref, out);
}